// LocationSensitiveAttention_46119358825192
// MI455X (gfx1250) — compile-verified
//
#include <hip/hip_runtime.h>
#include <hip/hip_bf16.h>
#include <math.h>

// Problem constants (match reference)
#define BB      64
#define TT_ALL  2048
#define LSTM    1024
#define ATTND   128
#define NFILT   32
#define KSZ     31
#define ENC     512
#define PAD     15

typedef __attribute__((ext_vector_type(2))) float v2f;
typedef __attribute__((ext_vector_type(4))) float f4;
typedef __attribute__((ext_vector_type(8))) float v8f;

// Fast tanh: CDNA5 has V_TANH_F32 as a hardware transcendental. Prefer the
// clang builtin if this toolchain declares it; otherwise emit the instruction
// directly. Raw asm bypasses compiler TRANS-hazard scheduling, so follow it
// with v_nop (ISA: 1 independent op required before a trans result is used).
__device__ __forceinline__ float fast_tanhf(float x)
{
#if __has_builtin(__builtin_amdgcn_tanhf)
    return __builtin_amdgcn_tanhf(x);
#else
    float r;
    asm volatile("v_tanh_f32 %0, %1\n\tv_nop" : "=v"(r) : "v"(x));
    return r;
#endif
}

// ---------------------------------------------------------------------------
// Kernel 1: pq[b][d] = sum_k decoder_q[b][k] * query_w[k][d]
// 64 blocks x 128 threads. W row access is wave-coalesced; q[k] is a scalar
// (uniform) load. ~17 MFLOPs total, negligible.
// ---------------------------------------------------------------------------
__global__ __launch_bounds__(128)
void pq_kernel(const float* __restrict__ q, const float* __restrict__ W,
               float* __restrict__ pq)
{
    const int b = blockIdx.x;
    const int d = threadIdx.x;
    const float* qb = q + b * LSTM;
    float acc = 0.f;
#pragma unroll 8
    for (int k = 0; k < LSTM; ++k)
        acc += qb[k] * W[k * ATTND + d];
    pq[b * ATTND + d] = acc;
}

// ---------------------------------------------------------------------------
// Kernel 2 (the big fused pass): per (b, 128-wide t-tile):
//   conv1d(attn_weights) -> locfeat[32][128] in LDS
//   loc2[tt][d] = locfeat^T @ loc_proj_w      via V_WMMA_F32_16X16X4_F32
//   e[t] = sum_d v_w[d] * tanh(pq[b][d] + loc2[tt][d] + pm[b][t][d])
// Streams processed_memory exactly once (64 MB, NT loads) -> ~3us floor.
// Block = 256 threads = 8 waves; wave w owns M-tile w (16 t-rows).
// Per-lane d-partials accumulate across N-tiles; one butterfly at the end.
// ---------------------------------------------------------------------------
__global__ __launch_bounds__(256)
void energies_kernel(const float* __restrict__ attn,     // [B,2,T]
                     const float* __restrict__ pm,       // [B,T,128]
                     const float* __restrict__ conv_w,   // [32,2,31]
                     const float* __restrict__ loc_proj, // [32,128]
                     const float* __restrict__ pq,       // [B,128]
                     const float* __restrict__ v_w,      // [128]
                     float* __restrict__ energies)       // [B,T]
{
    __shared__ float sAttn[2][128 + 2 * PAD + 2];  // halo tile of attn_weights
    __shared__ float sLoc[NFILT][136];             // conv output, pad=8 -> no bank clash

    const int b   = blockIdx.x;
    const int t0  = blockIdx.y * 128;
    const int tid = threadIdx.x;

    // --- stage attn_weights slice [t0-15, t0+142] with zero padding ---------
    for (int j = tid; j < 2 * (128 + 2 * PAD); j += 256) {
        const int c = j / (128 + 2 * PAD);
        const int x = j % (128 + 2 * PAD);
        const int g = t0 - PAD + x;
        float v = 0.f;
        if (g >= 0 && g < TT_ALL)
            v = attn[((size_t)b * 2 + c) * TT_ALL + g];
        sAttn[c][x] = v;
    }
    __syncthreads();

    // --- conv1d: locfeat[f][tt], 31-tap, 2 input channels -------------------
    // Within a wave f is uniform -> conv_w reads become scalar (L2-resident).
    for (int idx = tid; idx < NFILT * 128; idx += 256) {
        const int f  = idx >> 7;
        const int tt = idx & 127;
        const float* w0 = conv_w + f * (2 * KSZ);
        float s = 0.f;
#pragma unroll
        for (int k = 0; k < KSZ; ++k) {
            s += sAttn[0][tt + k] * w0[k];
            s += sAttn[1][tt + k] * w0[KSZ + k];
        }
        sLoc[f][tt] = s;
    }
    __syncthreads();

    // --- WMMA loc projection + fused tanh/v-dot -----------------------------
    const int wave   = tid >> 5;   // M-tile (16 t-rows each)
    const int lane   = tid & 31;
    const int lane16 = lane & 15;
    const int hi     = lane >> 4;  // 0: lanes 0-15, 1: lanes 16-31

    const float* pmB = pm + ((size_t)b * TT_ALL + t0) * ATTND;
    const float* pqB = pq + b * ATTND;

    float esum[8];
#pragma unroll
    for (int r = 0; r < 8; ++r) esum[r] = 0.f;

#pragma unroll
    for (int nt = 0; nt < 8; ++nt) {         // N-tiles over attn_dim (8x16 = 128)
        const int d   = nt * 16 + lane16;
        const float vw  = v_w[d];
        const float pqv = pqB[d];

        v8f acc = {};                         // C = 0
#pragma unroll
        for (int ks = 0; ks < 8; ++ks) {      // K-steps over 32 filters (8x4)
            const int fa = ks * 4 + hi * 2;   // lanes 0-15: K=0,1 ; 16-31: K=2,3
            v2f a, bmat;
            a.x    = sLoc[fa    ][wave * 16 + lane16];      // A: [tt][f] tile
            a.y    = sLoc[fa + 1][wave * 16 + lane16];
            bmat.x = loc_proj[(fa    ) * ATTND + d];        // B: [f][d] tile
            bmat.y = loc_proj[(fa + 1) * ATTND + d];
            acc = __builtin_amdgcn_wmma_f32_16x16x4_f32(
                      false, a, false, bmat, (short)0, acc, false, false);
        }

        // D layout: acc[r] is element (M = r + 8*hi, N = lane16).
        // Accumulate this lane's d-column partial; reduce across lanes later.
#pragma unroll
        for (int r = 0; r < 8; ++r) {
            const int m   = r + 8 * hi;
            const float pmv = __builtin_nontemporal_load(
                                  &pmB[(size_t)(wave * 16 + m) * ATTND + d]);
            esum[r] += vw * fast_tanhf(pqv + acc[r] + pmv);
        }
    }

    // One butterfly per r over the 16 N-lanes of this half (masks <= 8 never
    // cross the 16-lane boundary).
#pragma unroll
    for (int r = 0; r < 8; ++r) {
        float e = esum[r];
#pragma unroll
        for (int off = 8; off >= 1; off >>= 1)
            e += __shfl_xor(e, off, 32);
        esum[r] = e;
    }

    // lane 0 writes t = wave*16 + r ; lane 16 writes t = wave*16 + 8 + r
    if (lane16 == 0) {
#pragma unroll
        for (int r = 0; r < 8; ++r)
            energies[(size_t)b * TT_ALL + t0 + wave * 16 + 8 * hi + r] = esum[r];
    }
}

// ---------------------------------------------------------------------------
// Kernel 3: in-place masked softmax over T for each b.
// ---------------------------------------------------------------------------
__global__ __launch_bounds__(256)
void softmax_kernel(float* __restrict__ w, const unsigned char* __restrict__ mask)
{
    __shared__ float red[256];
    const int b   = blockIdx.x;
    const int tid = threadIdx.x;
    float vals[8];
    float mx = -INFINITY;
#pragma unroll
    for (int i = 0; i < 8; ++i) {
        const int t = tid + i * 256;
        float e = w[(size_t)b * TT_ALL + t];
        if (mask[(size_t)b * TT_ALL + t]) e = -INFINITY;
        vals[i] = e;
        mx = fmaxf(mx, e);
    }
    red[tid] = mx; __syncthreads();
    for (int s = 128; s > 0; s >>= 1) {
        if (tid < s) red[tid] = fmaxf(red[tid], red[tid + s]);
        __syncthreads();
    }
    mx = red[0]; __syncthreads();

    float sum = 0.f;
#pragma unroll
    for (int i = 0; i < 8; ++i) { vals[i] = expf(vals[i] - mx); sum += vals[i]; }
    red[tid] = sum; __syncthreads();
    for (int s = 128; s > 0; s >>= 1) {
        if (tid < s) red[tid] += red[tid + s];
        __syncthreads();
    }
    const float inv = 1.f / red[0];
#pragma unroll
    for (int i = 0; i < 8; ++i)
        w[(size_t)b * TT_ALL + tid + i * 256] = vals[i] * inv;
}

// ---------------------------------------------------------------------------
// Kernel 4: context partials. Grid (b, chunk of 128 t). Block 128 threads,
// one float4 of enc-dim each -> every t-step reads 2 KB contiguous, NT-hinted
// (memory is 256 MB > 192 MB L2 and read exactly once: bypass L2 retention).
// ---------------------------------------------------------------------------
#define CCHUNK 128
#define NCHUNK (TT_ALL / CCHUNK)   // 16

__global__ __launch_bounds__(128)
void context_partial(const float* __restrict__ memory, const float* __restrict__ w,
                     float* __restrict__ part)   // [NCHUNK][B][ENC]
{
    const int b   = blockIdx.x;
    const int ch  = blockIdx.y;
    const int tid = threadIdx.x;
    const float* memB = memory + ((size_t)b * TT_ALL + (size_t)ch * CCHUNK) * ENC;
    const float* wB   = w + (size_t)b * TT_ALL + ch * CCHUNK;
    f4 acc = {0.f, 0.f, 0.f, 0.f};
#pragma unroll 4
    for (int t = 0; t < CCHUNK; ++t) {
        const float wt = wB[t];                                  // scalar load
        const f4 m = __builtin_nontemporal_load(
                         (const f4*)(memB + (size_t)t * ENC) + tid);
        acc.x += wt * m.x; acc.y += wt * m.y;
        acc.z += wt * m.z; acc.w += wt * m.w;
    }
    ((f4*)(part + ((size_t)ch * BB + b) * ENC))[tid] = acc;
}

__global__ __launch_bounds__(256)
void context_reduce(const float* __restrict__ part, float* __restrict__ out)
{
    const int i = blockIdx.x * blockDim.x + threadIdx.x;   // over B*ENC
    float s = 0.f;
#pragma unroll
    for (int c = 0; c < NCHUNK; ++c)
        s += part[(size_t)c * BB * ENC + i];
    out[i] = s;
}

// ---------------------------------------------------------------------------
extern "C" void kernel_launch(void* const* d_in, const int* in_sizes, int n_in,
                              void* d_out, int out_size, void* d_ws, size_t ws_size,
                              hipStream_t stream)
{
    const float*         decoder_q = (const float*)d_in[0];
    const float*         memory    = (const float*)d_in[1];
    const float*         pm        = (const float*)d_in[2];
    const float*         attn      = (const float*)d_in[3];
    const unsigned char* mask      = (const unsigned char*)d_in[4];
    const float*         conv_w    = (const float*)d_in[5];
    const float*         loc_proj  = (const float*)d_in[6];
    const float*         query_w   = (const float*)d_in[7];
    const float*         v_w       = (const float*)d_in[8];

    float* out_ctx = (float*)d_out;                 // [B, ENC]
    float* out_w   = (float*)d_out + BB * ENC;      // [B, T] (energies in-place -> w)

    float* pq   = (float*)d_ws;                     // B*ATTND floats      (32 KB)
    float* part = (float*)d_ws + BB * ATTND;        // NCHUNK*B*ENC floats (2 MB)

    pq_kernel      <<<BB, 128, 0, stream>>>(decoder_q, query_w, pq);
    energies_kernel<<<dim3(BB, TT_ALL / 128), 256, 0, stream>>>(
        attn, pm, conv_w, loc_proj, pq, v_w, out_w);
    softmax_kernel <<<BB, 256, 0, stream>>>(out_w, mask);
    context_partial<<<dim3(BB, NCHUNK), 128, 0, stream>>>(memory, out_w, part);
    context_reduce <<<(BB * ENC) / 256, 256, 0, stream>>>(part, out_ctx);
}